// QuanvolutionFusion_65481071405655
// MI455X (gfx1250) — compile-verified
//
#include <hip/hip_runtime.h>
#include <math.h>

typedef __attribute__((ext_vector_type(2))) float v2f;
typedef __attribute__((ext_vector_type(8))) float v8f;

#define TILE_S 16
#define NPATCH 196
#define NFEAT  784
#define FEAT_STRIDE 788          // pad: s*788 mod 64 distinct for s=0..15 -> conflict-free A loads
#define NPAIR  392               // NFEAT/2 (K handled as pairs; one WMMA consumes 2 pairs)
#define BLOCK  128

// dynamic shared layout (in floats)
#define OFF_FEAT 0
#define SZ_FEAT  (TILE_S * FEAT_STRIDE)     // 12608
#define OFF_B    (OFF_FEAT + SZ_FEAT)
#define SZ_B     (NPAIR * 32)               // 12544 : sB[(j*16+n)*2+kk] = fc_w'[2j+kk][n]
#define OFF_RED  (OFF_B + SZ_B)
#define SZ_RED   (4 * 256)                  // per-wave partial C tiles
#define OFF_LOG  (OFF_RED + SZ_RED)
#define SZ_LOG   256                        // 16x16 logits tile
#define SMEM_FLOATS (OFF_LOG + SZ_LOG)
#define SMEM_BYTES  (SMEM_FLOATS * 4)

// RY(2*acos(c)) on qubit `bit` of a 16-amplitude real statevector (registers only)
__device__ __forceinline__ void ry16(float st[16], int bit, float c, float s) {
#pragma unroll
  for (int i = 0; i < 16; ++i) {
    if (!(i & bit)) {
      int j = i | bit;
      float a0 = st[i], a1 = st[j];
      st[i] = c * a0 - s * a1;
      st[j] = s * a0 + c * a1;
    }
  }
}

// CNOT(control=cbit, target=tbit): pure register permutation after full unroll
__device__ __forceinline__ void cnot16(float st[16], int cbit, int tbit) {
#pragma unroll
  for (int i = 0; i < 16; ++i) {
    if ((i & cbit) && !(i & tbit)) {
      int j = i | tbit;
      float t = st[i]; st[i] = st[j]; st[j] = t;
    }
  }
}

__global__ __launch_bounds__(BLOCK)
void quanv_fused_kernel(const float* __restrict__ x,       // [B,1,28,28]
                        const float* __restrict__ conv_w,  // [4,1,2,2]
                        const float* __restrict__ conv_b,  // [4]
                        const float* __restrict__ theta,   // [3,4]
                        const float* __restrict__ alpha_p, // [1]
                        const float* __restrict__ fc_w,    // [10,784]
                        const float* __restrict__ fc_b,    // [10]
                        float* __restrict__ out)           // [B,10]
{
  extern __shared__ float smem[];
  float* sFeat = smem + OFF_FEAT;
  float* sB    = smem + OFF_B;
  float* sRed  = smem + OFF_RED;
  float* sLog  = smem + OFF_LOG;

  const int tid = threadIdx.x;
  const int sampleBase = blockIdx.x * TILE_S;

  // ---- stage fc_w^T into LDS in the permuted feature order k' = p*4 + c ----
  // sB[(j*16+n)*2+kk] = fc_w[n][c*196+p] where k'=2j+kk, c=k'&3, p=k'>>2
  for (int e = tid; e < NPAIR * 32; e += BLOCK) {
    int kk = e & 1;
    int n  = (e >> 1) & 15;
    int j  = e >> 5;
    int kp = 2 * j + kk;
    int c  = kp & 3;
    int p  = kp >> 2;
    sB[e] = (n < 10) ? fc_w[n * NFEAT + c * NPATCH + p] : 0.0f;
  }

  // ---- uniform parameters into registers ----
  float cw[16], cb[4];
#pragma unroll
  for (int i = 0; i < 16; ++i) cw[i] = conv_w[i];
#pragma unroll
  for (int i = 0; i < 4; ++i) cb[i] = conv_b[i];
  float thc[12], ths[12];                 // half-angle sincos of theta (hoisted)
#pragma unroll
  for (int i = 0; i < 12; ++i) __sincosf(theta[i] * 0.5f, &ths[i], &thc[i]);
  const float a  = 1.0f / (1.0f + __expf(-alpha_p[0]));  // sigmoid(alpha)
  const float ia = 1.0f - a;

  // ---- phase 1: per-patch conv + 4-qubit statevector sim -> blended feat in LDS ----
  for (int t = tid; t < TILE_S * NPATCH; t += BLOCK) {
    int s  = t / NPATCH;
    int p  = t - s * NPATCH;
    int pi = p / 14, pj = p - pi * 14;
    const float* xp = x + (size_t)(sampleBase + s) * 784 + (2 * pi) * 28 + 2 * pj;
    float p00 = xp[0], p01 = xp[1], p10 = xp[28], p11 = xp[29];

    float conv[4];
#pragma unroll
    for (int c = 0; c < 4; ++c)
      conv[c] = fmaf(cw[c*4+0], p00, fmaf(cw[c*4+1], p01,
                fmaf(cw[c*4+2], p10, fmaf(cw[c*4+3], p11, cb[c]))));

    // input angles = patch * pi/2 ; RY uses half-angle -> patch * pi/4
    float inv[4] = {p00, p01, p10, p11};
    float ci[4], si[4];
#pragma unroll
    for (int k = 0; k < 4; ++k)
      __sincosf(inv[k] * 0.78539816339744830962f, &si[k], &ci[k]);

    float st[16];
#pragma unroll
    for (int i = 0; i < 16; ++i) st[i] = 0.0f;
    st[0] = 1.0f;
    // wire w <-> bit (8>>w); wire 0 is most-significant axis
    ry16(st, 8, ci[0], si[0]);
    ry16(st, 4, ci[1], si[1]);
    ry16(st, 2, ci[2], si[2]);
    ry16(st, 1, ci[3], si[3]);
#pragma unroll
    for (int l = 0; l < 3; ++l) {
      cnot16(st, 8, 4);   // CNOT(0,1)
      cnot16(st, 4, 2);   // CNOT(1,2)
      cnot16(st, 2, 1);   // CNOT(2,3)
      cnot16(st, 1, 8);   // CNOT(3,0)
      ry16(st, 8, thc[l*4+0], ths[l*4+0]);
      ry16(st, 4, thc[l*4+1], ths[l*4+1]);
      ry16(st, 2, thc[l*4+2], ths[l*4+2]);
      ry16(st, 1, thc[l*4+3], ths[l*4+3]);
    }
    float e0 = 0.f, e1 = 0.f, e2 = 0.f, e3 = 0.f;
#pragma unroll
    for (int i = 0; i < 16; ++i) {
      float pr = st[i] * st[i];
      e0 += (i & 8) ? -pr : pr;
      e1 += (i & 4) ? -pr : pr;
      e2 += (i & 2) ? -pr : pr;
      e3 += (i & 1) ? -pr : pr;
    }
    float ev[4] = {e0, e1, e2, e3};

    float* fp = sFeat + s * FEAT_STRIDE + p * 4;   // contiguous b128, conflict-free
#pragma unroll
    for (int c = 0; c < 4; ++c)
      fp[c] = a * conv[c] + ia * ev[c];
  }

  __syncthreads();

  // ---- phase 2: FC head with V_WMMA_F32_16X16X4_F32, K split across 4 waves ----
  const int wave = tid >> 5;
  const int lane = tid & 31;
  const int grp  = lane >> 4;   // K half (pairs j0 / j0+1)
  const int mrow = lane & 15;   // A: sample row ; B: class column

  v8f acc = {0.f, 0.f, 0.f, 0.f, 0.f, 0.f, 0.f, 0.f};
  const float* featRow = sFeat + mrow * FEAT_STRIDE;
  int j0 = wave * (NPAIR / 4);                 // 98 pairs per wave
  for (int it = 0; it < NPAIR / 8; ++it, j0 += 2) {   // 49 WMMA per wave
    int jj = j0 + grp;
    v2f afrag, bfrag;
    afrag.x = featRow[2 * jj + 0];             // A: M=mrow, K=2*grp+0
    afrag.y = featRow[2 * jj + 1];             // A: M=mrow, K=2*grp+1
    const float* bp = sB + (jj * 16 + mrow) * 2;
    bfrag.x = bp[0];                           // B: K=2*grp+0, N=mrow
    bfrag.y = bp[1];                           // B: K=2*grp+1, N=mrow
    acc = __builtin_amdgcn_wmma_f32_16x16x4_f32(
        /*neg_a=*/false, afrag, /*neg_b=*/false, bfrag,
        /*c_mod=*/(short)0, acc, /*reuse_a=*/false, /*reuse_b=*/false);
  }

  // stash per-wave partial C tiles
  float* rp = sRed + wave * 256 + lane * 8;
#pragma unroll
  for (int vg = 0; vg < 8; ++vg) rp[vg] = acc[vg];
  __syncthreads();

  // ---- reduce the 4 partial tiles, add bias ----
  for (int e = tid; e < 256; e += BLOCK) {
    float sum = sRed[e] + sRed[e + 256] + sRed[e + 512] + sRed[e + 768];
    int lane_e = e >> 3;
    int vg     = e & 7;
    int m = vg + ((lane_e & 16) ? 8 : 0);      // C/D layout: vgpr v -> rows v, v+8
    int n = lane_e & 15;
    sLog[m * 16 + n] = sum + ((n < 10) ? fc_b[n] : 0.0f);
  }
  __syncthreads();

  // ---- log_softmax over 10 classes, one thread per sample ----
  if (tid < TILE_S) {
    const int s = tid;
    float l[10];
    float mx = -1e30f;
#pragma unroll
    for (int n = 0; n < 10; ++n) { l[n] = sLog[s * 16 + n]; mx = fmaxf(mx, l[n]); }
    float se = 0.0f;
#pragma unroll
    for (int n = 0; n < 10; ++n) se += expf(l[n] - mx);
    float lse = logf(se) + mx;
    float* op = out + (size_t)(sampleBase + s) * 10;
#pragma unroll
    for (int n = 0; n < 10; ++n) op[n] = l[n] - lse;
  }
}

extern "C" void kernel_launch(void* const* d_in, const int* in_sizes, int n_in,
                              void* d_out, int out_size, void* d_ws, size_t ws_size,
                              hipStream_t stream) {
  (void)n_in; (void)out_size; (void)d_ws; (void)ws_size;
  const float* x      = (const float*)d_in[0];
  const float* conv_w = (const float*)d_in[1];
  const float* conv_b = (const float*)d_in[2];
  const float* theta  = (const float*)d_in[3];
  const float* alpha  = (const float*)d_in[4];
  const float* fc_w   = (const float*)d_in[5];
  const float* fc_b   = (const float*)d_in[6];
  float* out = (float*)d_out;

  const int B = in_sizes[0] / 784;            // 8192
  const int grid = B / TILE_S;                // 512 workgroups

  // allow >64KB dynamic LDS (gfx1250: up to 320KB per workgroup)
  hipFuncSetAttribute(reinterpret_cast<const void*>(quanv_fused_kernel),
                      hipFuncAttributeMaxDynamicSharedMemorySize, SMEM_BYTES);

  quanv_fused_kernel<<<grid, BLOCK, SMEM_BYTES, stream>>>(
      x, conv_w, conv_b, theta, alpha, fc_w, fc_b, out);
}